// BayesianEncoder_33328946217349
// MI455X (gfx1250) — compile-verified
//
#include <hip/hip_runtime.h>
#include <stdint.h>

// ---------------- problem constants ----------------
#define BATCH   200
#define BPAD    256          // 64 float4 per feature row -> 2x b128 per lane per column
#define NLAYER  13
#define NNZ_TOT 1070638
#define OUT_TOT 89064
#define FOUT    469

typedef __attribute__((ext_vector_type(2))) float v2f;
typedef __attribute__((ext_vector_type(4))) float v4f;
typedef __attribute__((ext_vector_type(8))) float v8f;

__constant__ int c_nnz_off[NLAYER + 1] = {
    0, 480000, 540000, 780000, 810000, 930000, 945000,
    1005000, 1012500, 1042500, 1046250, 1061258, 1063134, 1070638};
__constant__ int c_out_off[NLAYER + 1] = {
    0, 30000, 45000, 60000, 67500, 75000, 78750,
    82500, 84375, 86250, 87188, 88126, 88595, 89064};

// ---------------- helpers ----------------
__device__ __forceinline__ uint32_t hash_u32(uint32_t x) {
  x ^= x >> 16; x *= 0x7feb352dU;
  x ^= x >> 15; x *= 0x846ca68bU;
  x ^= x >> 16;
  return x;
}

// ---------------- kernels ----------------
__global__ void k_zero_i32(int* __restrict__ p, int n) {
  int i = blockIdx.x * blockDim.x + threadIdx.x;
  if (i < n) p[i] = 0;
}

// x (200 x F, batch-major) -> h (F x BPAD, feature-major), tiled LDS transpose
__global__ void k_transpose_in(const float* __restrict__ x, float* __restrict__ h, int F) {
  __shared__ float tile[32][33];
  int f0 = blockIdx.x * 32, b0 = blockIdx.y * 32;
  int tx = threadIdx.x, ty = threadIdx.y;  // block (32,8)
  for (int i = 0; i < 32; i += 8) {
    int b = b0 + ty + i, f = f0 + tx;
    tile[ty + i][tx] = (b < BATCH && f < F) ? x[(size_t)b * F + f] : 0.0f;
  }
  __syncthreads();
  for (int i = 0; i < 32; i += 8) {
    int f = f0 + ty + i, b = b0 + tx;
    if (f < F && b < BATCH) h[(size_t)f * BPAD + b] = tile[tx][ty + i];
  }
}

__global__ void k_edge_count(const int* __restrict__ edges, int* __restrict__ counts) {
  int e = blockIdx.x * blockDim.x + threadIdx.x;
  if (e >= NNZ_TOT) return;
  int dst = edges[NNZ_TOT + e];
  int li = 0;
#pragma unroll
  for (int k = 1; k < NLAYER; ++k) li += (e >= c_nnz_off[k]);
  atomicAdd(&counts[c_out_off[li] + dst], 1);
}

// single-block exclusive scan over OUT_TOT ints (1024 threads, Hillis-Steele per chunk)
__global__ void k_scan(const int* __restrict__ counts, int* __restrict__ row_start,
                       int* __restrict__ cursor) {
  __shared__ int sh[1024];
  __shared__ int s_carry;
  int t = threadIdx.x;
  if (t == 0) s_carry = 0;
  __syncthreads();
  for (int base = 0; base < OUT_TOT; base += 1024) {
    int v = (base + t < OUT_TOT) ? counts[base + t] : 0;
    sh[t] = v;
    __syncthreads();
    for (int off = 1; off < 1024; off <<= 1) {
      int add = (t >= off) ? sh[t - off] : 0;
      __syncthreads();
      sh[t] += add;
      __syncthreads();
    }
    int incl = sh[t];
    int carry = s_carry;
    if (base + t < OUT_TOT) {
      int ex = carry + incl - v;
      row_start[base + t] = ex;
      cursor[base + t] = ex;
    }
    __syncthreads();
    if (t == 1023) s_carry = carry + incl;
    __syncthreads();
  }
  if (t == 0) row_start[OUT_TOT] = s_carry;
}

__global__ void k_edge_scatter(const int* __restrict__ edges, int* __restrict__ cursor,
                               int* __restrict__ perm) {
  int e = blockIdx.x * blockDim.x + threadIdx.x;
  if (e >= NNZ_TOT) return;
  int dst = edges[NNZ_TOT + e];
  int li = 0;
#pragma unroll
  for (int k = 1; k < NLAYER; ++k) li += (e >= c_nnz_off[k]);
  int pos = atomicAdd(&cursor[c_out_off[li] + dst], 1);
  perm[pos] = e;
}

// one wave per output row; 32 lanes x 2 float4 cover the 256-slot batch column.
// Gathered columns are 1KB contiguous -> two global_load_b128 per lane per edge.
__global__ void k_spmm(const float* __restrict__ hin, float* __restrict__ hout,
                       const int* __restrict__ row_start, const int* __restrict__ perm,
                       const int* __restrict__ src_all, const float* __restrict__ w_mu,
                       const float* __restrict__ bias, int bias_off, int row_off, int rows) {
  int wave = (blockIdx.x * blockDim.x + threadIdx.x) >> 5;
  int lane = threadIdx.x & 31;
  if (wave >= rows) return;
  int e0 = row_start[row_off + wave];
  int e1 = row_start[row_off + wave + 1];
  v4f acc0 = {};
  v4f acc1 = {};
  for (int eb = e0; eb < e1; eb += 32) {
    int e = eb + lane;
    int sj = 0;
    float w = 0.0f;
    if (e < e1) {
      int pe = perm[e];
      sj = src_all[pe];
      w = w_mu[pe];
      __builtin_prefetch(hin + (size_t)sj * BPAD + 4 * lane, 0, 1);  // global_prefetch_b8
    }
    int cnt = min(32, e1 - eb);
    for (int k = 0; k < cnt; ++k) {
      int s = __shfl(sj, k, 32);
      float wk = __shfl(w, k, 32);
      const v4f* hp = (const v4f*)(hin + (size_t)s * BPAD);
      acc0 += wk * hp[lane];
      acc1 += wk * hp[32 + lane];
    }
  }
  float bj = bias[bias_off + wave];
  float* op = hout + (size_t)wave * BPAD;
  *(v4f*)(op + 4 * lane) = acc0 + bj;
  if (lane < 18) *(v4f*)(op + 128 + 4 * lane) = acc1 + bj;  // batches 128..199
}

// BN statistics via V_WMMA_F32_16X16X4_F32: colsum = ones(16x4) x H(4x16), exact fp32.
// One wave handles a 16-feature tile; K loop walks 200 batches 4 at a time.
// Pad lanes clamp to row F-1 so every load is an unconditional global_load_b64;
// their (garbage) sums are simply never written back.
__global__ void k_bn_stats(const float* __restrict__ h, float* __restrict__ mean,
                           float* __restrict__ rsig, int F) {
  int lane = threadIdx.x & 31;
  int wave = (blockIdx.x * blockDim.x + threadIdx.x) >> 5;
  int ntiles = (F + 15) >> 4;
  if (wave >= ntiles) return;
  int f0 = wave << 4;
  int fl = f0 + (lane & 15);
  bool valid = (fl < F);
  int fcl = valid ? fl : (F - 1);
  int bsub = (lane >> 4) * 2;  // B layout: lanes 0-15 -> K=0,1 ; lanes 16-31 -> K=2,3
  v2f ones;
  ones.x = 1.0f;
  ones.y = 1.0f;
  v8f accS = {};
  v8f accQ = {};
  const float* hp = h + (size_t)fcl * BPAD + bsub;
  for (int b0 = 0; b0 < BATCH; b0 += 4) {
    v2f bv = *(const v2f*)(hp + b0);           // unconditional global_load_b64
    v2f bq;
    bq.x = bv.x * bv.x;
    bq.y = bv.y * bv.y;
    accS = __builtin_amdgcn_wmma_f32_16x16x4_f32(false, ones, false, bv, (short)0, accS,
                                                 false, false);
    accQ = __builtin_amdgcn_wmma_f32_16x16x4_f32(false, ones, false, bq, (short)0, accQ,
                                                 false, false);
  }
  // A==ones -> every D row holds the column sums; D VGPR0, N = lane&15
  if (lane < 16 && valid) {
    float s = accS[0];
    float q = accQ[0];
    float m = s * (1.0f / BATCH);
    float var = q * (1.0f / BATCH) - m * m;
    mean[fl] = m;
    rsig[fl] = rsqrtf(var + 1e-5f);
  }
}

// normalize * gamma + beta, relu, dropout (deterministic hash) — in place
__global__ void k_bn_apply(float* __restrict__ h, const float* __restrict__ mean,
                           const float* __restrict__ rsig, const float* __restrict__ gamma,
                           const float* __restrict__ beta, int gb_off, int stage) {
  int f = blockIdx.x;
  int b = threadIdx.x;
  if (b >= BATCH) return;
  size_t idx = (size_t)f * BPAD + b;
  float v = (h[idx] - mean[f]) * rsig[f];
  v = v * gamma[gb_off + f] + beta[gb_off + f];
  v = fmaxf(v, 0.0f);
  uint32_t hsh = hash_u32(hash_u32(0x9E3779B9u * (100u + stage)) ^
                          (uint32_t)(f * 0x85EBCA77u) ^ (uint32_t)(b * 0xC2B2AE3Du));
  float u = (hsh >> 8) * (1.0f / 16777216.0f);
  v = (u < 0.9f) ? v * (1.0f / 0.9f) : 0.0f;
  h[idx] = v;
}

// final stage-6 BN (no affine), write batch-major output
__global__ void k_final_out(const float* __restrict__ h, const float* __restrict__ mean,
                            const float* __restrict__ rsig, float* __restrict__ out) {
  int f = blockIdx.x;
  int b = threadIdx.x;
  if (b >= BATCH) return;
  float v = (h[(size_t)f * BPAD + b] - mean[f]) * rsig[f];
  out[(size_t)b * FOUT + f] = v;
}

// KL = sum 0.5*(exp(2 ls) + mu^2 - 1 - 2 ls); deterministic two-pass reduction
__global__ void k_kl_partial(const float* __restrict__ w_mu, const float* __restrict__ w_ls,
                             float* __restrict__ part) {
  __shared__ float sh[256];
  float a = 0.0f;
  for (int e = blockIdx.x * 256 + threadIdx.x; e < NNZ_TOT; e += 256 * 256) {
    float mu = w_mu[e], ls = w_ls[e];
    a += 0.5f * (expf(2.0f * ls) + mu * mu - 1.0f - 2.0f * ls);
  }
  sh[threadIdx.x] = a;
  __syncthreads();
  for (int o = 128; o > 0; o >>= 1) {
    if (threadIdx.x < o) sh[threadIdx.x] += sh[threadIdx.x + o];
    __syncthreads();
  }
  if (threadIdx.x == 0) part[blockIdx.x] = sh[0];
}

__global__ void k_kl_final(const float* __restrict__ part, float* __restrict__ out) {
  __shared__ float sh[256];
  sh[threadIdx.x] = part[threadIdx.x];
  __syncthreads();
  for (int o = 128; o > 0; o >>= 1) {
    if (threadIdx.x < o) sh[threadIdx.x] += sh[threadIdx.x + o];
    __syncthreads();
  }
  if (threadIdx.x == 0) out[(size_t)BATCH * FOUT] = sh[0];
}

// ---------------- host launcher ----------------
extern "C" void kernel_launch(void* const* d_in, const int* in_sizes, int n_in,
                              void* d_out, int out_size, void* d_ws, size_t ws_size,
                              hipStream_t stream) {
  (void)in_sizes; (void)n_in; (void)out_size; (void)ws_size;
  const float* x      = (const float*)d_in[0];
  const int*   edges  = (const int*)d_in[1];   // [2, NNZ_TOT]: srcs then dsts
  const float* w_mu   = (const float*)d_in[2];
  const float* w_ls   = (const float*)d_in[3];
  const float* bias   = (const float*)d_in[4];
  const float* gamma  = (const float*)d_in[5];
  const float* beta   = (const float*)d_in[6];
  float* out = (float*)d_out;

  static const int h_out_off[NLAYER + 1] = {
      0, 30000, 45000, 60000, 67500, 75000, 78750,
      82500, 84375, 86250, 87188, 88126, 88595, 89064};
  static const int h_out_size[NLAYER] = {30000, 15000, 15000, 7500, 7500, 3750, 3750,
                                         1875, 1875, 938, 938, 469, 469};
  static const int h_bn_off[7] = {0, 30000, 45000, 52500, 56250, 58125, 59063};

  // workspace carve-up (256B aligned)
  uint8_t* base = (uint8_t*)d_ws;
  size_t off = 0;
  auto take = [&](size_t bytes) -> void* {
    void* p = base + off;
    off = (off + bytes + 255) & ~(size_t)255;
    return p;
  };
  float* buf0     = (float*)take((size_t)30000 * BPAD * sizeof(float));
  float* buf1     = (float*)take((size_t)30000 * BPAD * sizeof(float));
  int*   counts   = (int*)take((size_t)OUT_TOT * sizeof(int));
  int*   row_st   = (int*)take((size_t)(OUT_TOT + 1) * sizeof(int));
  int*   cursor   = (int*)take((size_t)OUT_TOT * sizeof(int));
  int*   perm     = (int*)take((size_t)NNZ_TOT * sizeof(int));
  float* mean     = (float*)take((size_t)30000 * sizeof(float));
  float* rsig     = (float*)take((size_t)30000 * sizeof(float));
  float* klpart   = (float*)take((size_t)256 * sizeof(float));

  // 1) CSR build
  k_zero_i32<<<(OUT_TOT + 255) / 256, 256, 0, stream>>>(counts, OUT_TOT);
  k_edge_count<<<(NNZ_TOT + 255) / 256, 256, 0, stream>>>(edges, counts);
  k_scan<<<1, 1024, 0, stream>>>(counts, row_st, cursor);
  k_edge_scatter<<<(NNZ_TOT + 255) / 256, 256, 0, stream>>>(edges, cursor, perm);

  // 2) transpose input to feature-major
  {
    dim3 g((30000 + 31) / 32, (BATCH + 31) / 32);
    dim3 b(32, 8);
    k_transpose_in<<<g, b, 0, stream>>>(x, buf0, 30000);
  }

  // 3) layer pipeline (ping-pong buf0/buf1; even layers end a stage -> BN on buf1)
  for (int li = 0; li < NLAYER; ++li) {
    const float* hin = (li % 2 == 0) ? buf0 : buf1;
    float* hout = (li % 2 == 0) ? buf1 : buf0;
    int rows = h_out_size[li];
    int blocks = (rows + 7) / 8;  // 8 waves (rows) per 256-thread block
    k_spmm<<<blocks, 256, 0, stream>>>(hin, hout, row_st, perm, edges, w_mu, bias,
                                       h_out_off[li], h_out_off[li], rows);
    if ((li & 1) == 0) {
      int stage = li / 2;
      int F = rows;
      int ntiles = (F + 15) / 16;
      k_bn_stats<<<(ntiles + 7) / 8, 256, 0, stream>>>(hout, mean, rsig, F);
      if (stage < 6) {
        k_bn_apply<<<F, 256, 0, stream>>>(hout, mean, rsig, gamma, beta,
                                          h_bn_off[stage], stage);
      } else {
        k_final_out<<<F, 256, 0, stream>>>(hout, mean, rsig, out);
      }
    }
  }

  // 4) KL scalar
  k_kl_partial<<<256, 256, 0, stream>>>(w_mu, w_ls, klpart);
  k_kl_final<<<1, 256, 0, stream>>>(klpart, out);
}